// GCNLayer_566935683471
// MI455X (gfx1250) — compile-verified
//
#include <hip/hip_runtime.h>

#define D 128
#define WLDS_STRIDE 132   // 128 + 4 pad floats -> 64-bank conflict-free b64 reads

typedef float v2f __attribute__((ext_vector_type(2)));
typedef float v8f __attribute__((ext_vector_type(8)));

// ---------------- Kernel 1: zero the aggregation buffer h ----------------
__global__ void zero_f4_kernel(float4* __restrict__ p, int n4) {
    int i = blockIdx.x * blockDim.x + threadIdx.x;
    if (i < n4) p[i] = make_float4(0.f, 0.f, 0.f, 0.f);
}

// ---------------- Kernel 2: COO SpMM scatter: h[row] += val * X[col] ------
// One wave32 per edge; each lane owns 4 consecutive features (float4 gather,
// 512B coalesced per wave). Scatter via hardware f32 atomics (L2-resident h).
__global__ __launch_bounds__(256)
void spmm_scatter_kernel(const float* __restrict__ X,
                         const int*   __restrict__ rows,
                         const int*   __restrict__ cols,
                         const float* __restrict__ vals,
                         float* __restrict__ h,
                         int nEdges) {
    int gid  = blockIdx.x * blockDim.x + threadIdx.x;
    int e    = gid >> 5;
    int lane = gid & 31;
    e = __builtin_amdgcn_readfirstlane(e);     // wave-uniform -> scalar loads
    if (e >= nEdges) return;

    int   r = rows[e];
    int   c = cols[e];
    float v = vals[e];

    const float4 xv = ((const float4*)(X + (size_t)c * D))[lane];
    float* hp = h + (size_t)r * D + lane * 4;
    unsafeAtomicAdd(hp + 0, v * xv.x);
    unsafeAtomicAdd(hp + 1, v * xv.y);
    unsafeAtomicAdd(hp + 2, v * xv.z);
    unsafeAtomicAdd(hp + 3, v * xv.w);
}

// ---------------- Kernel 3: out = h @ W^T + b via V_WMMA_F32_16X16X4_F32 --
// Block = 256 threads = 8 waves; each wave computes a 16x128 output stripe
// as 8 tiles of 16x16, K-loop of 32 x (16x16x4 f32 WMMA). W (64KB) is staged
// in LDS (padded stride -> conflict-free) so WGP$ is free to stream h.
//
// ISA VGPR layouts (CDNA5 ISA):
//   A (16x4, MxK): lane L -> M = L%16, a[v] = A[M][2*(L/16) + v]
//   B (4x16, KxN): lane L -> N = L%16, b[v] = B[2*(L/16) + v][N]
//   C/D (16x16):   lane L, vgpr v -> (M = v + 8*(L/16), N = L%16)
__global__ __launch_bounds__(256)
void gemm_wmma_kernel(const float* __restrict__ h,
                      const float* __restrict__ W,     // [128 out, 128 in]
                      const float* __restrict__ bias,  // [128]
                      float* __restrict__ out,
                      int nNodes) {
    __shared__ float Wl[D * WLDS_STRIDE];               // 66 KB

    const int wave = threadIdx.x >> 5;   // 0..7
    const int lane = threadIdx.x & 31;
    const int l15  = lane & 15;
    const int hi   = lane >> 4;          // 0 or 1
    const int rowBase = blockIdx.x * 128 + wave * 16;

    // Stage W into LDS: 4096 float4 loads spread over 256 threads.
    for (int i = threadIdx.x; i < (D * D) / 4; i += 256) {
        float4 w4 = ((const float4*)W)[i];
        int flat = i * 4;
        int o = flat >> 7;        // / 128
        int k = flat & 127;       // % 128
        *(float4*)&Wl[o * WLDS_STRIDE + k] = w4;
    }
    __syncthreads();

    // A pointer for this lane's row (clamped for the ragged tail; stores are
    // predicated below, so clamped rows only produce discarded garbage).
    int arow = rowBase + l15;
    if (arow >= nNodes) arow = nNodes - 1;
    const float* aptr = h + (size_t)arow * D + 2 * hi;

    v8f acc[8];
#pragma unroll
    for (int nt = 0; nt < 8; ++nt) acc[nt] = (v8f){0.f,0.f,0.f,0.f,0.f,0.f,0.f,0.f};

    // B element: B[k_local][o] = W^T tile = W[o][k_global], read from LDS.
    const float* bbase = Wl + (size_t)l15 * WLDS_STRIDE + 2 * hi;

    for (int ks = 0; ks < 32; ++ks) {
        const int k = ks * 4;
        const v2f a = *(const v2f*)(aptr + k);
#pragma unroll
        for (int nt = 0; nt < 8; ++nt) {
            const v2f b2 = *(const v2f*)(bbase + nt * 16 * WLDS_STRIDE + k);
            acc[nt] = __builtin_amdgcn_wmma_f32_16x16x4_f32(
                /*neg_a=*/false, a, /*neg_b=*/false, b2,
                /*c_mod=*/(short)0, acc[nt],
                /*reuse_a=*/false, /*reuse_b=*/false);
        }
    }

    // Store D + bias; each (nt, v) store is two 64B coalesced runs and the
    // 8 nt-tiles fully cover each 512B row segment. NT stores: out is
    // write-once, keep it from evicting h/X/W from L2.
#pragma unroll
    for (int nt = 0; nt < 8; ++nt) {
        const float bv = bias[nt * 16 + l15];
#pragma unroll
        for (int v = 0; v < 8; ++v) {
            const int r = rowBase + v + 8 * hi;
            if (r < nNodes)
                __builtin_nontemporal_store(acc[nt][v] + bv,
                                            &out[(size_t)r * D + nt * 16 + l15]);
        }
    }
}

extern "C" void kernel_launch(void* const* d_in, const int* in_sizes, int n_in,
                              void* d_out, int out_size, void* d_ws, size_t ws_size,
                              hipStream_t stream) {
    const float* X    = (const float*)d_in[0];
    const int*   rows = (const int*)  d_in[1];
    const int*   cols = (const int*)  d_in[2];
    const float* vals = (const float*)d_in[3];
    const float* W    = (const float*)d_in[4];
    const float* bias = (const float*)d_in[5];
    float*       out  = (float*)d_out;
    float*       h    = (float*)d_ws;          // [nNodes, 128] f32 scratch

    const int nNodes = in_sizes[0] / D;
    const int nEdges = in_sizes[1];

    // 1) h = 0
    {
        const int n4 = nNodes * (D / 4);
        zero_f4_kernel<<<(n4 + 255) / 256, 256, 0, stream>>>((float4*)h, n4);
    }
    // 2) h[row] += val * X[col]   (one wave per edge)
    {
        const long long threads = (long long)nEdges * 32;
        const int blocks = (int)((threads + 255) / 256);
        spmm_scatter_kernel<<<blocks, 256, 0, stream>>>(X, rows, cols, vals, h, nEdges);
    }
    // 3) out = h @ W^T + b
    {
        const int blocks = (nNodes + 127) / 128;   // 128 rows per block
        gemm_wmma_kernel<<<blocks, 256, 0, stream>>>(h, W, bias, out, nNodes);
    }
}